// EdgeConv2d_31945966748194
// MI455X (gfx1250) — compile-verified
//
#include <hip/hip_runtime.h>
#include <stdint.h>

typedef __attribute__((ext_vector_type(2))) float v2f;
typedef __attribute__((ext_vector_type(8))) float v8f;

#define N_PTS 16384
#define C_IN 64
#define K_NB 16
#define C_OUT 64
#define B_SZ 4
#define COL_STRIDE 68   // 16 columns x 68 floats (padded) -> conflict-free b64 LDS reads
#define W_STRIDE 68

#if __has_builtin(__builtin_amdgcn_global_load_async_to_lds_b64)
#define USE_ASYNC_LDS 1
#else
#define USE_ASYNC_LDS 0
#endif

// Exact parameter types per hipcc diagnostic:
//   param0: __attribute__((vector_size(8))) int __device__*   (addrspace(1))
//   param1: same pointee in __shared__ (addrspace(3))
typedef int v2i __attribute__((vector_size(8)));
typedef __attribute__((address_space(1))) v2i* gbl_v2i_p;
typedef __attribute__((address_space(3))) v2i* lds_v2i_p;

// Copy 8 bytes global -> per-wave LDS staging. On gfx1250 use the async
// global->LDS pipe (ASYNCcnt); otherwise stage through VGPRs.
__device__ __forceinline__ void gather8(const float* __restrict__ gsrc, float* ldst) {
#if USE_ASYNC_LDS
  __builtin_amdgcn_global_load_async_to_lds_b64(
      (gbl_v2i_p)(uintptr_t)gsrc,
      (lds_v2i_p)(unsigned)(uintptr_t)ldst, 0, 0);
#else
  *(v2f*)ldst = *(const v2f*)gsrc;
#endif
}

__device__ __forceinline__ void gather_fence_pre() {
  // WAR: make sure prior DS reads of the staging buffer have landed before
  // the async pipe (or new stores) overwrite it.
  asm volatile("s_wait_dscnt 0x0" ::: "memory");
}

__device__ __forceinline__ void gather_fence_post() {
#if USE_ASYNC_LDS
  asm volatile("s_wait_asynccnt 0x0" ::: "memory");
#else
  asm volatile("" ::: "memory");  // DS pipe is in-order per wave; just stop reordering
#endif
}

// ---------------------------------------------------------------------------
// Pass 0: transpose x (B,C,N) -> xt (B,N,C) so gathered columns are contiguous.
// Also zero the per-channel stats accumulators (required every call: the graph
// is replayed without re-initializing workspace).
// ---------------------------------------------------------------------------
__global__ __launch_bounds__(256) void transpose_x_kernel(const float* __restrict__ x,
                                                          float* __restrict__ xt,
                                                          float* __restrict__ stats) {
  __shared__ float tile[64][65];
  const int b = blockIdx.y;
  const int n0 = blockIdx.x * 64;
  for (int i = threadIdx.x; i < 64 * 64; i += 256) {
    int c = i >> 6, n = i & 63;                       // consecutive threads -> consecutive n (coalesced)
    tile[c][n] = x[(b * C_IN + c) * N_PTS + n0 + n];
  }
  __syncthreads();
  for (int i = threadIdx.x; i < 64 * 64; i += 256) {
    int n = i >> 6, c = i & 63;                       // consecutive threads -> consecutive c (coalesced)
    xt[(b * N_PTS + n0 + n) * C_IN + c] = tile[c][n];
  }
  if (blockIdx.x == 0 && blockIdx.y == 0 && threadIdx.x < 128) stats[threadIdx.x] = 0.0f;
}

// ---------------------------------------------------------------------------
// Fold mean/var/gamma/beta into per-channel scale/shift.
// ---------------------------------------------------------------------------
__global__ __launch_bounds__(64) void finalize_stats_kernel(const float* __restrict__ stats,
                                                            const float* __restrict__ gamma,
                                                            const float* __restrict__ beta,
                                                            float* __restrict__ scsh) {
  const int o = threadIdx.x;
  const float inv_m = 1.0f / (float)(B_SZ * N_PTS * K_NB);
  float mean = stats[o] * inv_m;
  float var  = stats[64 + o] * inv_m - mean * mean;
  float rstd = rsqrtf(var + 1e-5f);
  float sc   = gamma[o] * rstd;
  scsh[o]      = sc;
  scsh[64 + o] = beta[o] - mean * sc;
}

// ---------------------------------------------------------------------------
// Main WMMA kernel. PHASE 1: accumulate per-channel sum / sum-of-squares of h.
// PHASE 2: recompute h, apply BN scale/shift + LeakyReLU, max over K, store.
//
// Wave-level tiling: each wave (32 lanes) owns 16 nodes.
//   t  = (W1+W2) @ Xc[16 nodes]            (one 64x16 tile set)
//   per node: S = W2 @ XJ[16 neighbors]    (one 64x16 tile set, W2 frags in VGPRs)
//   h[:,k] = t[:,node] - S[:,k]
// Fragment layouts (V_WMMA_F32_16X16X4_F32):
//   A 16x4 : lane<16 -> row=lane, K={0,1}; lane>=16 -> row=lane-16, K={2,3}
//   B 4x16 : lane<16 -> col=lane, K={0,1}; lane>=16 -> col=lane-16, K={2,3}
//   C/D    : reg v, lane<16 -> (row v,   col lane); lane>=16 -> (row v+8, col lane-16)
// ---------------------------------------------------------------------------
template <int PHASE>
__global__ __launch_bounds__(128) void edgeconv_wmma_kernel(
    const float* __restrict__ xt, const int* __restrict__ eidx, const float* __restrict__ W,
    float* __restrict__ stats, const float* __restrict__ scsh, float* __restrict__ out) {
  __shared__ float lds_u[C_OUT * W_STRIDE];       // W1+W2 (used once -> stays in LDS)
  __shared__ float lds_w2[C_OUT * W_STRIDE];      // W2 (hoisted to VGPR frags below)
  __shared__ float lds_cols[4][16 * COL_STRIDE];  // per-wave staging of 16 feature columns

  const int tid  = threadIdx.x;
  const int wave = tid >> 5;
  const int lane = tid & 31;
  const int hf   = lane >> 4;       // which 16-lane half
  const int lc   = lane & 15;
  const int coff = hf * 2;          // K-offset within a 4-wide WMMA step: 0 or 2
  const int b    = blockIdx.y;
  const int n0   = blockIdx.x * 64 + wave * 16;
  float* cols = &lds_cols[wave][0];

  // Stage W into LDS in both forms.
  for (int i = tid; i < C_OUT * C_IN; i += 128) {
    int o = i >> 6, c = i & 63;
    float w2 = W[o * (2 * C_IN) + C_IN + c];
    lds_w2[o * W_STRIDE + c] = w2;
    lds_u[o * W_STRIDE + c]  = W[o * (2 * C_IN) + c] + w2;
  }
  __syncthreads();

  // Hoist the full W2 A-fragment set into VGPRs (4 co-blocks x 16 K-steps x v2f
  // = 128 VGPRs; wave32 budget is 1024). Removes 64 DS loads per node tile.
  v2f wfrag[4][16];
#pragma unroll
  for (int cob = 0; cob < 4; ++cob)
#pragma unroll
    for (int kk = 0; kk < 16; ++kk)
      wfrag[cob][kk] = *(const v2f*)&lds_w2[(cob * 16 + lc) * W_STRIDE + kk * 4 + coff];

  // --- load this wave's 16 self-feature columns into LDS -------------------
  gather_fence_pre();
  for (int col = 0; col < 16; ++col)
    gather8(&xt[(b * N_PTS + n0 + col) * C_IN + lane * 2], &cols[col * COL_STRIDE + lane * 2]);
  gather_fence_post();

  // --- t = (W1+W2) @ Xc -----------------------------------------------------
  v8f tacc[4] = {};
#pragma unroll
  for (int kk = 0; kk < 16; ++kk) {
    v2f bfrag = *(const v2f*)&cols[lc * COL_STRIDE + kk * 4 + coff];
#pragma unroll
    for (int cob = 0; cob < 4; ++cob) {
      v2f afrag = *(const v2f*)&lds_u[(cob * 16 + lc) * W_STRIDE + kk * 4 + coff];
      tacc[cob] = __builtin_amdgcn_wmma_f32_16x16x4_f32(false, afrag, false, bfrag,
                                                        (short)0, tacc[cob], false, false);
    }
  }

  v8f ssum[4] = {};
  v8f ssq[4]  = {};
  float sc[4][8], sh[4][8];
  if (PHASE == 2) {
#pragma unroll
    for (int cob = 0; cob < 4; ++cob)
#pragma unroll
      for (int v = 0; v < 8; ++v) {
        int o = cob * 16 + v + hf * 8;
        sc[cob][v] = scsh[o];
        sh[cob][v] = scsh[64 + o];
      }
  }

  // --- per-node: S = W2 @ XJ, combine, reduce ------------------------------
  for (int j = 0; j < 16; ++j) {
    const int n = n0 + j;
    const int m = eidx[(b * N_PTS + n) * K_NB + lc];  // lanes 0..15 carry the 16 neighbor ids

    // gather 16 neighbor columns (each 64 contiguous floats in xt) into LDS
    gather_fence_pre();
    for (int col = 0; col < 16; ++col) {
      int mm = __shfl(m, col, 32);
      gather8(&xt[(b * N_PTS + mm) * C_IN + lane * 2], &cols[col * COL_STRIDE + lane * 2]);
    }
    gather_fence_post();

    v8f acc[4] = {};
#pragma unroll
    for (int kk = 0; kk < 16; ++kk) {
      v2f bfrag = *(const v2f*)&cols[lc * COL_STRIDE + kk * 4 + coff];
#pragma unroll
      for (int cob = 0; cob < 4; ++cob) {
        acc[cob] = __builtin_amdgcn_wmma_f32_16x16x4_f32(false, wfrag[cob][kk], false, bfrag,
                                                         (short)0, acc[cob], false, false);
      }
    }

    const int src = j + hf * 16;  // lane holding t[:,node j] for this half's rows
#pragma unroll
    for (int cob = 0; cob < 4; ++cob) {
      if (PHASE == 1) {
#pragma unroll
        for (int v = 0; v < 8; ++v) {
          float tj = __shfl(tacc[cob][v], src, 32);
          float h  = tj - acc[cob][v];
          ssum[cob][v] += h;
          ssq[cob][v]  += h * h;
        }
      } else {
        float r[8];
#pragma unroll
        for (int v = 0; v < 8; ++v) {
          float tj  = __shfl(tacc[cob][v], src, 32);
          float h   = tj - acc[cob][v];
          float val = sc[cob][v] * h + sh[cob][v];
          r[v] = (val >= 0.0f) ? val : 0.2f * val;
        }
        // max over the 16 neighbor columns (held in the 16 lanes of each half)
#pragma unroll
        for (int mask = 1; mask < 16; mask <<= 1)
#pragma unroll
          for (int v = 0; v < 8; ++v) r[v] = fmaxf(r[v], __shfl_xor(r[v], mask, 32));
        if (lc == 0) {
#pragma unroll
          for (int v = 0; v < 8; ++v) {
            int o = cob * 16 + v + hf * 8;
            out[(b * C_OUT + o) * N_PTS + n] = r[v];
          }
        }
      }
    }
  }

  if (PHASE == 1) {
    // per-wave cross-lane reduction (within each 16-lane half), then one atomic
#pragma unroll
    for (int cob = 0; cob < 4; ++cob)
#pragma unroll
      for (int v = 0; v < 8; ++v) {
        float s = ssum[cob][v];
        float q = ssq[cob][v];
#pragma unroll
        for (int mask = 1; mask < 16; mask <<= 1) {
          s += __shfl_xor(s, mask, 32);
          q += __shfl_xor(q, mask, 32);
        }
        if (lc == 0) {
          int o = cob * 16 + v + hf * 8;
          atomicAdd(&stats[o], s);
          atomicAdd(&stats[64 + o], q);
        }
      }
  }
}

// ---------------------------------------------------------------------------
extern "C" void kernel_launch(void* const* d_in, const int* in_sizes, int n_in,
                              void* d_out, int out_size, void* d_ws, size_t ws_size,
                              hipStream_t stream) {
  const float* x     = (const float*)d_in[0];
  const int*   eidx  = (const int*)d_in[1];
  const float* W     = (const float*)d_in[2];
  const float* gamma = (const float*)d_in[3];
  const float* beta  = (const float*)d_in[4];
  float* out = (float*)d_out;

  float* xt    = (float*)d_ws;                       // B*N*C floats = 16 MB
  float* stats = xt + (size_t)B_SZ * N_PTS * C_IN;   // sum[64], sumsq[64]
  float* scsh  = stats + 128;                        // scale[64], shift[64]

  dim3 tgrid(N_PTS / 64, B_SZ);
  transpose_x_kernel<<<tgrid, 256, 0, stream>>>(x, xt, stats);

  dim3 mgrid(N_PTS / 64, B_SZ);  // 4 waves/block x 16 nodes/wave = 64 nodes/block
  edgeconv_wmma_kernel<1><<<mgrid, 128, 0, stream>>>(xt, eidx, W, stats, scsh, out);

  finalize_stats_kernel<<<1, 64, 0, stream>>>(stats, gamma, beta, scsh);

  edgeconv_wmma_kernel<2><<<mgrid, 128, 0, stream>>>(xt, eidx, W, stats, scsh, out);
}